// MPHeteNewHead_14448269984048
// MI455X (gfx1250) — compile-verified
//
#include <hip/hip_runtime.h>
#include <hip/hip_bf16.h>

#define ND 100000
#define NT 512
#define DIMV 128
#define NE 500000
#define NP 500000

typedef __attribute__((ext_vector_type(2))) float v2f;
typedef __attribute__((ext_vector_type(8))) float v8f;
typedef __attribute__((ext_vector_type(4))) int v4i;

#if defined(__has_builtin)
#if __has_builtin(__builtin_amdgcn_global_load_async_to_lds_b128)
#define HAVE_ASYNC_LDS 1
#endif
#endif

__device__ __forceinline__ void atomAddG(float* p, float v) {
    __hip_atomic_fetch_add(p, v, __ATOMIC_RELAXED, __HIP_MEMORY_SCOPE_AGENT);
}
__device__ __forceinline__ void atomAddL(float* p, float v) {
    __hip_atomic_fetch_add(p, v, __ATOMIC_RELAXED, __HIP_MEMORY_SCOPE_WORKGROUP);
}

__device__ __forceinline__ float waveReduceSum32(float s) {
#pragma unroll
    for (int o = 16; o; o >>= 1) s += __shfl_xor(s, o, 32);
    return s;
}

// ---------------------------------------------------------------------------
// Row-wise L2 normalization: one wave per 128-float row, float4 per lane.
// ---------------------------------------------------------------------------
__global__ __launch_bounds__(256) void l2norm_rows(const float* __restrict__ in,
                                                   float* __restrict__ out, int nRows) {
    int wid = blockIdx.x * 8 + (threadIdx.x >> 5);
    int lane = threadIdx.x & 31;
    if (wid >= nRows) return;
    const float4* r = (const float4*)(in + (size_t)wid * DIMV);
    float4 v = r[lane];
    float s = v.x * v.x + v.y * v.y + v.z * v.z + v.w * v.w;
    s = waveReduceSum32(s);
    float inv = 1.0f / fmaxf(sqrtf(s), 1e-12f);
    float4 o4 = make_float4(v.x * inv, v.y * inv, v.z * inv, v.w * inv);
    ((float4*)(out + (size_t)wid * DIMV))[lane] = o4;
}

// ---------------------------------------------------------------------------
// WMMA f32 GEMM: Out(32 rows x 128) = A(32x128) @ W(128x128) [+ M + bias]
// [+ relu] [+ row l2norm].
// Block = 256 threads = 8 waves; wave w owns columns [16w, 16w+16) and computes
// TWO stacked 16x16 tiles (rows 0-15 and 16-31) that SHARE one set of B loads,
// halving the L2 traffic for W.  A tile is staged into LDS via
// GLOBAL_LOAD_ASYNC_TO_LDS_B128 (ASYNCcnt) when available.
// ---------------------------------------------------------------------------
__global__ __launch_bounds__(256) void gemm32(const float* __restrict__ A,
                                              const float* __restrict__ W,
                                              const float* __restrict__ Madd,
                                              const float* __restrict__ bias,
                                              float* __restrict__ Out,
                                              int doRelu, int doNorm) {
    __shared__ float As[32 * DIMV];  // 16 KB
    __shared__ float Os[32 * DIMV];  // 16 KB

    const int r0 = blockIdx.x * 32;

    // Stage the 32x128 A tile into LDS: 4096 floats = 256 threads x 4 float4.
#ifdef HAVE_ASYNC_LDS
    {
        const float4* gsrc = (const float4*)(A + (size_t)r0 * DIMV);
#pragma unroll
        for (int k = 0; k < 4; ++k) {
            int i = threadIdx.x + k * 256;
            __builtin_amdgcn_global_load_async_to_lds_b128(
                (__attribute__((address_space(1))) v4i*)((void*)(gsrc + i)),
                (__attribute__((address_space(3))) v4i*)((void*)(((float4*)As) + i)),
                0, 0);
        }
        asm volatile("s_wait_asynccnt 0" ::: "memory");
    }
#else
#pragma unroll
    for (int k = 0; k < 4; ++k) {
        int i = threadIdx.x + k * 256;
        ((float4*)As)[i] = ((const float4*)(A + (size_t)r0 * DIMV))[i];
    }
#endif
    __syncthreads();

    const int wave = threadIdx.x >> 5;
    const int lane = threadIdx.x & 31;
    const int mrow = lane & 15;
    const int koff = (lane < 16) ? 0 : 2;   // per ISA 16x4 f32 A layout
    const int n    = (wave << 4) + (lane & 15);

    v8f acc0 = {};
    v8f acc1 = {};
#pragma unroll 8
    for (int kb = 0; kb < DIMV; kb += 4) {
        v2f b;
        b.x = W[(size_t)(kb + koff) * DIMV + n];
        b.y = W[(size_t)(kb + koff + 1) * DIMV + n];
        v2f a0, a1;
        a0.x = As[mrow * DIMV + kb + koff];
        a0.y = As[mrow * DIMV + kb + koff + 1];
        a1.x = As[(16 + mrow) * DIMV + kb + koff];
        a1.y = As[(16 + mrow) * DIMV + kb + koff + 1];
        acc0 = __builtin_amdgcn_wmma_f32_16x16x4_f32(
            false, a0, false, b, (short)0, acc0, false, false);
        acc1 = __builtin_amdgcn_wmma_f32_16x16x4_f32(
            false, a1, false, b, (short)0, acc1, false, false);
    }

    if (!Madd && !doNorm) {
        // Plain store: D layout VGPR r -> M = r + 8*(lane>=16)
#pragma unroll
        for (int r = 0; r < 8; ++r) {
            int Mr = r + ((lane < 16) ? 0 : 8);
            Out[(size_t)(r0 + Mr) * DIMV + n] = acc0[r];
            Out[(size_t)(r0 + 16 + Mr) * DIMV + n] = acc1[r];
        }
        return;
    }

    // Epilogue: + message + bias, optional relu, stage to LDS for row norms.
#pragma unroll
    for (int r = 0; r < 8; ++r) {
        int Mr = r + ((lane < 16) ? 0 : 8);
        float v0 = acc0[r];
        float v1 = acc1[r];
        if (Madd) {
            v0 += Madd[(size_t)(r0 + Mr) * DIMV + n];
            v1 += Madd[(size_t)(r0 + 16 + Mr) * DIMV + n];
        }
        if (bias) {
            float bv = bias[n];
            v0 += bv;
            v1 += bv;
        }
        if (doRelu) {
            v0 = fmaxf(v0, 0.0f);
            v1 = fmaxf(v1, 0.0f);
        }
        Os[Mr * DIMV + n] = v0;
        Os[(16 + Mr) * DIMV + n] = v1;
    }
    __syncthreads();

    // Row l2norm: 32 rows, 8 threads per row, 16 columns (4 float4) per thread.
    const int row = threadIdx.x >> 3;
    const int sub = threadIdx.x & 7;
    const int base = row * DIMV + sub * 16;
    float4 xv[4];
    float s = 0.0f;
#pragma unroll
    for (int j = 0; j < 4; ++j) {
        xv[j] = ((const float4*)(Os + base))[j];
        s += xv[j].x * xv[j].x + xv[j].y * xv[j].y + xv[j].z * xv[j].z + xv[j].w * xv[j].w;
    }
#pragma unroll
    for (int o = 4; o; o >>= 1) s += __shfl_xor(s, o, 32);  // stays in 8-lane group
    float inv = 1.0f / fmaxf(sqrtf(s), 1e-12f);
    float4* dst = (float4*)(Out + (size_t)(r0 + row) * DIMV + sub * 16);
#pragma unroll
    for (int j = 0; j < 4; ++j)
        dst[j] = make_float4(xv[j].x * inv, xv[j].y * inv, xv[j].z * inv, xv[j].w * inv);
}

// ---------------------------------------------------------------------------
// Edge kernel: for each edge e (data node a = ed[e], task t = et[e], w = ev[e])
//   m_t[t] += y_d[a] * w   (segment-sum into 512-row table -> LDS accumulate)
//   m_d[a] += y_t[t] * w   (scatter over 100K rows -> direct f32 atomics)
// grid = (chunks, 4 dim-slices of 32); one wave per edge per slice,
// lane = dim within slice -> 128B coalesced gathers.
// ---------------------------------------------------------------------------
__global__ __launch_bounds__(256) void edge_kernel(const float* __restrict__ yd,
                                                   const float* __restrict__ yt,
                                                   const float* __restrict__ ev,
                                                   const int* __restrict__ ed,
                                                   const int* __restrict__ et,
                                                   float* __restrict__ md,
                                                   float* __restrict__ mt,
                                                   int nEdge) {
    __shared__ float mts[NT * 32];  // 64 KB: full task table, 32-dim slice

    const int dimBase = blockIdx.y * 32;
    for (int i = threadIdx.x; i < NT * 32; i += 256) mts[i] = 0.0f;
    __syncthreads();

    const int lane = threadIdx.x & 31;
    const int wid = blockIdx.x * 8 + (threadIdx.x >> 5);
    const int stride = gridDim.x * 8;

    for (int e = wid; e < nEdge; e += stride) {
        int a = ed[e];
        int t = et[e];
        float w = ev[e];
        float v = yd[(size_t)a * DIMV + dimBase + lane];
        atomAddL(&mts[t * 32 + lane], v * w);
        float u = yt[(size_t)t * DIMV + dimBase + lane];
        atomAddG(&md[(size_t)a * DIMV + dimBase + lane], u * w);
    }
    __syncthreads();

    for (int i = threadIdx.x; i < NT * 32; i += 256) {
        float v = mts[i];
        if (v != 0.0f)
            atomAddG(&mt[(size_t)(i >> 5) * DIMV + dimBase + (i & 31)], v);
    }
}

// ---------------------------------------------------------------------------
// Prediction: pred[i] = dot(x_d[idx0[i]], x_t[idx1[i]]). One wave per pred,
// float4 per lane, shfl reduction.
// ---------------------------------------------------------------------------
__global__ __launch_bounds__(256) void pred_kernel(const float* __restrict__ xd,
                                                   const float* __restrict__ xt,
                                                   const int* __restrict__ pidx,
                                                   float* __restrict__ out, int nPred) {
    int wid = blockIdx.x * 8 + (threadIdx.x >> 5);
    int lane = threadIdx.x & 31;
    if (wid >= nPred) return;
    int a = pidx[wid];
    int b = pidx[nPred + wid];
    float4 va = ((const float4*)(xd + (size_t)a * DIMV))[lane];
    float4 vb = ((const float4*)(xt + (size_t)b * DIMV))[lane];
    float s = va.x * vb.x + va.y * vb.y + va.z * vb.z + va.w * vb.w;
    s = waveReduceSum32(s);
    if (lane == 0) out[wid] = s;
}

// ---------------------------------------------------------------------------
extern "C" void kernel_launch(void* const* d_in, const int* in_sizes, int n_in,
                              void* d_out, int out_size, void* d_ws, size_t ws_size,
                              hipStream_t stream) {
    const float* gf   = (const float*)d_in[0];
    const float* te   = (const float*)d_in[1];
    const float* ev   = (const float*)d_in[2];
    const float* Wsd  = (const float*)d_in[3];
    const float* Wst  = (const float*)d_in[4];
    const float* Wd2t = (const float*)d_in[5];
    const float* Wt2d = (const float*)d_in[6];
    const float* bd   = (const float*)d_in[7];
    const float* bt   = (const float*)d_in[8];
    const int*   ed   = (const int*)d_in[9];
    const int*   et   = (const int*)d_in[10];
    const int*   pidx = (const int*)d_in[11];
    float* out = (float*)d_out;

    char* ws = (char*)d_ws;
    const size_t szd = (size_t)ND * DIMV * sizeof(float);  // 51.2 MB
    const size_t szt = (size_t)NT * DIMV * sizeof(float);  // 256 KB
    float* xd[2] = {(float*)ws, (float*)(ws + szd)};
    float* yd = (float*)(ws + 2 * szd);
    float* md = (float*)(ws + 3 * szd);
    char* p = ws + 4 * szd;
    float* xt[2] = {(float*)p, (float*)(p + szt)};
    float* yt = (float*)(p + 2 * szt);
    float* mt = (float*)(p + 3 * szt);

    l2norm_rows<<<ND / 8, 256, 0, stream>>>(gf, xd[0], ND);
    l2norm_rows<<<NT / 8, 256, 0, stream>>>(te, xt[0], NT);

    int cur = 0;
    for (int l = 0; l < 2; ++l) {
        const size_t wOff = (size_t)l * DIMV * DIMV;
        // Message projections
        gemm32<<<ND / 32, 256, 0, stream>>>(xd[cur], Wd2t + wOff, nullptr, nullptr, yd, 0, 0);
        gemm32<<<NT / 32, 256, 0, stream>>>(xt[cur], Wt2d + wOff, nullptr, nullptr, yt, 0, 0);
        // Zero message accumulators
        (void)hipMemsetAsync(md, 0, szd, stream);
        (void)hipMemsetAsync(mt, 0, szt, stream);
        // Gather/scatter over edges
        edge_kernel<<<dim3(160, 4), 256, 0, stream>>>(yd, yt, ev, ed, et, md, mt, NE);
        // Self transform + message + bias (+relu on layer 0) + l2norm
        gemm32<<<ND / 32, 256, 0, stream>>>(xd[cur], Wsd + wOff, md, bd + (size_t)l * DIMV,
                                            xd[1 - cur], (l == 0) ? 1 : 0, 1);
        gemm32<<<NT / 32, 256, 0, stream>>>(xt[cur], Wst + wOff, mt, bt + (size_t)l * DIMV,
                                            xt[1 - cur], (l == 0) ? 1 : 0, 1);
        cur = 1 - cur;
    }

    pred_kernel<<<NP / 8, 256, 0, stream>>>(xd[cur], xt[cur], pidx, out, NP);
}